// PEPS_model_1391569404358
// MI455X (gfx1250) — compile-verified
//
#include <hip/hip_runtime.h>

typedef float v2f __attribute__((ext_vector_type(2)));
typedef float v8f __attribute__((ext_vector_type(8)));

#define LX 6
#define LY 6
#define PD 2
#define DD 4
#define NSITE 36
#define ENV7 16384   /* 4^7 floats, peak env size */
#define ENV6 4096    /* 4^6 floats, between-column env */

// -------- prep: build zero-padded site matrices --------
// Msite[((site*2+s)*16 + k)*16 + n], k = a*4+c (GEMM K), n = d*4+b (GEMM N)
// padded to zero on boundary-sliced bonds so every site is a full 16x16.
__global__ void peps_prep_kernel(const float* __restrict__ peps,
                                 float* __restrict__ Msite) {
    int idx = blockIdx.x * blockDim.x + threadIdx.x;
    if (idx >= NSITE * PD * 256) return;
    int n    = idx & 15;
    int k    = (idx >> 4) & 15;
    int s    = (idx >> 8) & 1;
    int site = idx >> 9;
    int i = site / LY, j = site % LY;
    int a = k >> 2, c = k & 3;
    int d = n >> 2, b = n & 3;
    float v = 0.0f;
    bool pad = (i == 0      && a > 0) || (i == LX - 1 && b > 0) ||
               (j == 0      && c > 0) || (j == LY - 1 && d > 0);
    if (!pad) {
        int p = ((((site * PD + s) * DD + a) * DD + b) * DD + c) * DD + d;
        v = peps[p];
    }
    Msite[idx] = v;
}

// -------- main: one block = one sample; env lives in LDS --------
__global__ void __launch_bounds__(256)
peps_amp_kernel(const int* __restrict__ x,
                const float* __restrict__ Msite,
                float* __restrict__ out) {
    extern __shared__ float env[];   // ENV7 floats = 64 KB

    const int tid    = threadIdx.x;
    const int sample = blockIdx.x;
    const int lane   = tid & 31;
    // wave id is uniform across the wave: pin it in an SGPR so the tile loop
    // compiles as a scalar loop (EXEC stays all-1s, as WMMA requires).
    const int wave   = __builtin_amdgcn_readfirstlane(tid >> 5);
    __builtin_assume(wave >= 0 && wave < 8);
    const int n      = lane & 15;    // GEMM col for B/C/D, GEMM row for A loads
    const int h      = lane >> 4;    // lane half

    // env <- one-hot (padded ones((1,)*6) boundary env)
    for (int q = tid; q < ENV7; q += 256) env[q] = 0.0f;
    __syncthreads();
    if (tid == 0) env[0] = 1.0f;
    __syncthreads();

    const int* xrow = x + sample * NSITE;

    for (int j = 0; j < LY; ++j) {
        for (int i = 0; i < LX; ++i) {
            const int site = i * LY + j;
            const int s    = xrow[site];
            const float* Mp = Msite + (site * PD + s) * 256;

            // B operand: 16x16 matrix as 4 K-chunks of 4x16.
            // VGPR v, half h -> K = 4*kc + 2*h + v, col N = n.
            v2f b0, b1, b2, b3;
            b0.x = Mp[(0  + 2 * h) * 16 + n];  b0.y = Mp[(1  + 2 * h) * 16 + n];
            b1.x = Mp[(4  + 2 * h) * 16 + n];  b1.y = Mp[(5  + 2 * h) * 16 + n];
            b2.x = Mp[(8  + 2 * h) * 16 + n];  b2.y = Mp[(9  + 2 * h) * 16 + n];
            b3.x = Mp[(12 + 2 * h) * 16 + n];  b3.y = Mp[(13 + 2 * h) * 16 + n];

            const int shift = 2 * (5 - i);     // post_size = 4^(5-i)
            const int psm   = (1 << shift) - 1;

            // 64 row-tiles of 16 env rows; 8 tiles per wave; in-place GEMM.
            #pragma unroll 2
            for (int t = wave; t < 64; t += 8) {
                // A operand: lane holds env row r = t*16 + n, K = 4kc+2h+{0,1}
                int r    = t * 16 + n;
                int post = r & psm;
                int base = ((r - post) << 4) + post;   // pre*16*ps + post
                v2f a0, a1, a2, a3;
                a0.x = env[base + ((0  + 2 * h) << shift)];
                a0.y = env[base + ((1  + 2 * h) << shift)];
                a1.x = env[base + ((4  + 2 * h) << shift)];
                a1.y = env[base + ((5  + 2 * h) << shift)];
                a2.x = env[base + ((8  + 2 * h) << shift)];
                a2.y = env[base + ((9  + 2 * h) << shift)];
                a3.x = env[base + ((12 + 2 * h) << shift)];
                a3.y = env[base + ((13 + 2 * h) << shift)];

                v8f acc = {};
                acc = __builtin_amdgcn_wmma_f32_16x16x4_f32(
                        false, a0, false, b0, (short)0, acc, false, false);
                acc = __builtin_amdgcn_wmma_f32_16x16x4_f32(
                        false, a1, false, b1, (short)0, acc, false, false);
                acc = __builtin_amdgcn_wmma_f32_16x16x4_f32(
                        false, a2, false, b2, (short)0, acc, false, false);
                acc = __builtin_amdgcn_wmma_f32_16x16x4_f32(
                        false, a3, false, b3, (short)0, acc, false, false);

                // store: lane holds col N=n of rows M = v + 8h of this tile
                #pragma unroll
                for (int v = 0; v < 8; ++v) {
                    int rs    = t * 16 + v + 8 * h;
                    int posts = rs & psm;
                    int bs    = ((rs - posts) << 4) + posts;
                    env[bs + (n << shift)] = acc[v];
                }
            }
            __syncthreads();
        }

        // column transition: env6[C] = env7[4*C] (b5=0 slice), zero the rest
        if (j < LY - 1) {
            float keep[16];
            #pragma unroll
            for (int q = 0; q < 16; ++q) keep[q] = env[4 * (tid + 256 * q)];
            __syncthreads();
            #pragma unroll
            for (int q = 0; q < 16; ++q) env[tid + 256 * q] = keep[q];
            for (int q = ENV6 + tid; q < ENV7; q += 256) env[q] = 0.0f;
            __syncthreads();
        }
    }

    if (tid == 0) out[sample] = env[0];
}

extern "C" void kernel_launch(void* const* d_in, const int* in_sizes, int n_in,
                              void* d_out, int out_size, void* d_ws, size_t ws_size,
                              hipStream_t stream) {
    const int*   x     = (const int*)d_in[0];    // (B, 36) int32
    const float* peps  = (const float*)d_in[1];  // (6,6,2,4,4,4,4) f32
    float*       out   = (float*)d_out;          // (B,) f32
    float*       Msite = (float*)d_ws;           // 36*2*256 floats = 73728 B

    const int nM = NSITE * PD * 256;             // 18432
    peps_prep_kernel<<<(nM + 255) / 256, 256, 0, stream>>>(peps, Msite);

    const int B = out_size;                      // 8192 samples
    peps_amp_kernel<<<B, 256, ENV7 * sizeof(float), stream>>>(x, Msite, out);
}